// SecondaryCapsuleLayer_1434519077397
// MI455X (gfx1250) — compile-verified
//
#include <hip/hip_runtime.h>
#include <hip/hip_bf16.h>
#include <math.h>

typedef float v2f __attribute__((ext_vector_type(2)));
typedef float v8f __attribute__((ext_vector_type(8)));

#define B_SZ     32
#define N_NODES  128
#define IN_CH    16
#define IN_DIM   32
#define OUT_CH   32
#define OUT_DIM  32
#define K_CAPS   (N_NODES * IN_CH)   // 2048 input capsules per (b, o)
#define U_STRIDE 33                  // pad: gcd(33,64)=1 -> conflict-free LDS
#define NUM_ITERS 3
#define EPS_SQ   1e-11f

// LDS floats: u[2048*33] + e[2048] + b[2048] + red[256] + vv[32] + scal[4]
#define SMEM_FLOATS (K_CAPS * U_STRIDE + K_CAPS + K_CAPS + 256 + 32 + 4)
#define SMEM_BYTES  (SMEM_FLOATS * sizeof(float))

__global__ __launch_bounds__(256) void capsule_route_kernel(
    const float* __restrict__ x,   // [B, N, IN_CH, IN_DIM]
    const float* __restrict__ W,   // [IN_CH, OUT_CH, IN_DIM, OUT_DIM]
    float* __restrict__ out)       // v_j [B*OUT_CH*OUT_DIM] then a_j [B*OUT_CH]
{
    extern __shared__ float smem[];
    float* u     = smem;                        // [K_CAPS][U_STRIDE]
    float* e_arr = u + K_CAPS * U_STRIDE;       // exp(b - max)
    float* b_arr = e_arr + K_CAPS;              // routing logits
    float* red   = b_arr + K_CAPS;              // [256] reduction scratch
    float* vv    = red + 256;                   // [32] v_j
    float* scal  = vv + 32;                     // [0] = a_j

    const int bo   = blockIdx.x;
    const int bb   = bo >> 5;        // batch index
    const int o    = bo & 31;        // output channel
    const int tid  = threadIdx.x;
    const int wave = tid >> 5;
    const int lane = tid & 31;
    const int half = lane >> 4;      // 0: lanes 0-15, 1: lanes 16-31
    const int lm   = lane & 15;

    // init routing logits
    for (int k = tid; k < K_CAPS; k += 256) b_arr[k] = 0.0f;

    // ---- Phase 1: u_hat[n*16+c][j] = sum_i x[b,n,c,i] * W[c,o,i,j] via WMMA f32 ----
    // Each wave owns 2 input channels c; B fragments for a channel stay register-
    // resident across all 8 M-tiles. Two N-tiles per job -> two independent
    // accumulator chains sharing each A fragment.
    for (int ci = 0; ci < 2; ++ci) {
        const int c = wave * 2 + ci;             // 0..15

        // Preload all B fragments for this channel: bf[kstep][nt]
        // B 4x16 f32 layout: lanes 0-15 -> rows i0,i0+1 ; lanes 16-31 -> i0+2,i0+3
        const float* wb = W + (((c * OUT_CH + o) << 5) << 5) + lm;
        v2f bf[8][2];
        #pragma unroll
        for (int ks = 0; ks < 8; ++ks) {
            const int i0 = 4 * ks + 2 * half;
            bf[ks][0].x = wb[i0 * OUT_DIM];
            bf[ks][0].y = wb[(i0 + 1) * OUT_DIM];
            bf[ks][1].x = wb[i0 * OUT_DIM + 16];
            bf[ks][1].y = wb[(i0 + 1) * OUT_DIM + 16];
        }

        for (int mt = 0; mt < 8; ++mt) {
            const int n = mt * 16 + lm;          // A row (per ISA layout, both halves)
            const float* xrow = x + (((bb * N_NODES + n) * IN_CH + c) << 5);

            v8f acc0 = {};                       // columns 0..15
            v8f acc1 = {};                       // columns 16..31
            #pragma unroll
            for (int ks = 0; ks < 8; ++ks) {
                // A 16x4 f32: lanes 0-15 -> K=k0,k0+1 ; lanes 16-31 -> K=k0+2,k0+3
                v2f av;
                av.x = xrow[4 * ks + 2 * half];
                av.y = xrow[4 * ks + 2 * half + 1];
                acc0 = __builtin_amdgcn_wmma_f32_16x16x4_f32(
                    false, av, false, bf[ks][0], (short)0, acc0, false, false);
                acc1 = __builtin_amdgcn_wmma_f32_16x16x4_f32(
                    false, av, false, bf[ks][1], (short)0, acc1, false, false);
            }
            // D 16x16 f32: VGPR r holds rows r (lanes 0-15) and r+8 (lanes 16-31)
            #pragma unroll
            for (int r = 0; r < 8; ++r) {
                const int m    = r + 8 * half;
                const int kcap = (mt * 16 + m) * IN_CH + c;
                u[kcap * U_STRIDE + lm]      = acc0[r];
                u[kcap * U_STRIDE + 16 + lm] = acc1[r];
            }
        }
    }
    __syncthreads();

    // ---- Phase 2: dynamic routing, 3 iterations, all in LDS ----
    const int jj = tid & 31;
    const int gg = tid >> 5;

    for (int it = 0; it < NUM_ITERS; ++it) {
        // softmax over k: max
        float lmax = -3.402823466e38f;
        for (int k = tid; k < K_CAPS; k += 256) lmax = fmaxf(lmax, b_arr[k]);
        red[tid] = lmax;
        __syncthreads();
        for (int s = 128; s > 0; s >>= 1) {
            if (tid < s) red[tid] = fmaxf(red[tid], red[tid + s]);
            __syncthreads();
        }
        const float bmax = red[0];
        __syncthreads();

        // exp + sum
        float lsum = 0.0f;
        for (int k = tid; k < K_CAPS; k += 256) {
            const float e = __expf(b_arr[k] - bmax);
            e_arr[k] = e;
            lsum += e;
        }
        red[tid] = lsum;
        __syncthreads();
        for (int s = 128; s > 0; s >>= 1) {
            if (tid < s) red[tid] += red[tid + s];
            __syncthreads();
        }
        const float inv_sum = 1.0f / red[0];
        __syncthreads();

        // s_j = (1/sum) * sum_k e_k * u[k][j]  (8 partial groups per j)
        float sp = 0.0f;
        for (int k = gg; k < K_CAPS; k += 8) sp += e_arr[k] * u[k * U_STRIDE + jj];
        red[tid] = sp;
        __syncthreads();

        if (tid < 32) {
            float sj = 0.0f;
            #pragma unroll
            for (int g = 0; g < 8; ++g) sj += red[g * 32 + jj];
            sj *= inv_sum;
            // squash: mag_sq via wave32 xor-reduction
            float msq = sj * sj;
            #pragma unroll
            for (int off = 16; off > 0; off >>= 1) msq += __shfl_xor(msq, off);
            const float aj = msq / (1.0f + msq);
            vv[jj] = aj * sj / (sqrtf(msq) + EPS_SQ);
            if (jj == 0) scal[0] = aj;
        }
        __syncthreads();

        // b_ij += <u_hat[k], v>   (skip on final iteration; result unused)
        if (it < NUM_ITERS - 1) {
            for (int k = tid; k < K_CAPS; k += 256) {
                const float* ur = &u[k * U_STRIDE];
                float dot = 0.0f;
                #pragma unroll
                for (int j2 = 0; j2 < OUT_DIM; ++j2) dot += ur[j2] * vv[j2];
                b_arr[k] += dot;
            }
            __syncthreads();
        }
    }

    // ---- Output: v_j [B,1,32,32] then a_j [B,1,32,1] ----
    if (tid < 32) {
        out[(bb * OUT_CH + o) * OUT_DIM + tid] = vv[tid];
        if (tid == 0) out[B_SZ * OUT_CH * OUT_DIM + bb * OUT_CH + o] = scal[0];
    }
}

extern "C" void kernel_launch(void* const* d_in, const int* in_sizes, int n_in,
                              void* d_out, int out_size, void* d_ws, size_t ws_size,
                              hipStream_t stream) {
    const float* x = (const float*)d_in[0];
    const float* W = (const float*)d_in[1];
    // d_in[2] = number_of_nodes (128), compile-time constant here
    float* out = (float*)d_out;

    capsule_route_kernel<<<B_SZ * OUT_CH, 256, SMEM_BYTES, stream>>>(x, W, out);
}